// FullTensorProduct_47012712022438
// MI455X (gfx1250) — compile-verified
//
#include <hip/hip_runtime.h>
#include <stdint.h>

// ===========================================================================
// MAIN KERNEL FIRST (so the disasm snippet shows the hot kernel's codegen:
// async global->LDS staging, s_wait_asynccnt, ds reads, scalarized CG math).
// ===========================================================================

#define USE_ASYNC_LDS 1

#if USE_ASYNC_LDS
__device__ __forceinline__ void async_load16(const float* gsrc, unsigned lds_byte) {
  unsigned long long ga = (unsigned long long)(size_t)gsrc;
  asm volatile("global_load_async_to_lds_b128 %0, %1, off"
               :: "v"(lds_byte), "v"(ga) : "memory");
}
#endif

// One path: y[i][k] = sum_j cg[i,j,k]*x2[j] (row-uniform -> scalar pipe), then
// acc[k] += y[i][k]*x1[u,i] per lane-owned u, contiguous d3-float store.
template <int D1, int D2, int D3, int UPL>
__device__ __forceinline__ void run_path(const float* __restrict__ cgp,
                                         const float (&x1r)[UPL][D1],
                                         const float* __restrict__ x2p,
                                         float* __restrict__ outp,
                                         const int lane) {
  float y[D1][D3];
#pragma unroll
  for (int i = 0; i < D1; ++i) {
#pragma unroll
    for (int k = 0; k < D3; ++k) {
      float s = cgp[(i * D2) * D3 + k] * x2p[0];
#pragma unroll
      for (int j = 1; j < D2; ++j) s = fmaf(cgp[(i * D2 + j) * D3 + k], x2p[j], s);
      y[i][k] = s;
    }
  }
#pragma unroll
  for (int t = 0; t < UPL; ++t) {
    const int u = lane + 32 * t;
    float acc[D3];
#pragma unroll
    for (int k = 0; k < D3; ++k) acc[k] = y[0][k] * x1r[t][0];
#pragma unroll
    for (int i = 1; i < D1; ++i)
#pragma unroll
      for (int k = 0; k < D3; ++k) acc[k] = fmaf(y[i][k], x1r[t][i], acc[k]);
#pragma unroll
    for (int k = 0; k < D3; ++k) outp[u * D3 + k] = acc[k];
  }
}

__global__ __launch_bounds__(256) void ftp_kernel(const float* __restrict__ in1,
                                                  const float* __restrict__ in2,
                                                  const float* __restrict__ cg,
                                                  float* __restrict__ out,
                                                  int nrows) {
  __shared__ __align__(16) float lds[8 * 832];   // 26.6 KB: one 3328B row slot per wave
  const int lane = (int)(threadIdx.x & 31u);
  const int wave = (int)(threadIdx.x >> 5u);
  int row = __builtin_amdgcn_readfirstlane((int)blockIdx.x * 8 + wave);
  if (row >= nrows) return;

  float* slot = &lds[wave * 832];
  const float* g1 = in1 + (size_t)row * 832;

  // ---- stage the in1 row into this wave's LDS slot (coalesced 16B/lane) ----
#if USE_ASYNC_LDS
  {
    const unsigned lbase = (unsigned)(size_t)(void*)slot;
#pragma unroll
    for (int c = 0; c < 6; ++c) {              // 6*32 = 192 chunks = 3072 B
      const int ci = c * 32 + lane;
      async_load16(g1 + ci * 4, lbase + (unsigned)(ci * 16));
    }
    if (lane < 16) {                           // tail: chunks 192..207
      const int ci = 192 + lane;
      async_load16(g1 + ci * 4, lbase + (unsigned)(ci * 16));
    }
    asm volatile("s_wait_asynccnt 0" ::: "memory");
  }
#else
  for (int ci = lane; ci < 208; ci += 32)
    *(float4*)(slot + ci * 4) = *(const float4*)(g1 + ci * 4);
#endif

  // ---- x2 row (9 floats, wave-uniform -> scalar loads) ----
  const float* g2 = in2 + (size_t)row * 9;
  float x2v[9];
#pragma unroll
  for (int j = 0; j < 9; ++j) x2v[j] = g2[j];

  float* orow = out + (size_t)row * 7488;

  { // -------- l1 = 0 : u = lane+32t, t<4, x1 at slot[u] --------
    float xr[4][1];
#pragma unroll
    for (int t = 0; t < 4; ++t) xr[t][0] = slot[lane + 32 * t];
    run_path<1, 1, 1, 4>(cg + 0,  xr, x2v + 0, orow + 0,    lane);  // 0x0->0
    run_path<1, 3, 3, 4>(cg + 1,  xr, x2v + 1, orow + 320,  lane);  // 0x1->1
    run_path<1, 5, 5, 4>(cg + 10, xr, x2v + 4, orow + 2240, lane);  // 0x2->2
  }
  { // -------- l1 = 1 : x1 at slot[128 + u*3 + i] --------
    float xr[4][3];
#pragma unroll
    for (int t = 0; t < 4; ++t)
#pragma unroll
      for (int i = 0; i < 3; ++i) xr[t][i] = slot[128 + (lane + 32 * t) * 3 + i];
    run_path<3, 1, 3, 4>(cg + 35,  xr, x2v + 0, orow + 704,  lane); // 1x0->1
    run_path<3, 3, 1, 4>(cg + 44,  xr, x2v + 1, orow + 128,  lane); // 1x1->0
    run_path<3, 3, 3, 4>(cg + 53,  xr, x2v + 1, orow + 1664, lane); // 1x1->1
    run_path<3, 3, 5, 4>(cg + 80,  xr, x2v + 1, orow + 2880, lane); // 1x1->2
    run_path<3, 5, 3, 4>(cg + 125, xr, x2v + 4, orow + 1088, lane); // 1x2->1
    run_path<3, 5, 5, 4>(cg + 170, xr, x2v + 4, orow + 4160, lane); // 1x2->2
    run_path<3, 5, 7, 4>(cg + 245, xr, x2v + 4, orow + 5120, lane); // 1x2->3
  }
  { // -------- l1 = 2 : u = lane+32t, t<2, x1 at slot[512 + u*5 + i] --------
    float xr[2][5];
#pragma unroll
    for (int t = 0; t < 2; ++t)
#pragma unroll
      for (int i = 0; i < 5; ++i) xr[t][i] = slot[512 + (lane + 32 * t) * 5 + i];
    run_path<5, 1, 5, 2>(cg + 350,  xr, x2v + 0, orow + 3520, lane); // 2x0->2
    run_path<5, 3, 3, 2>(cg + 375,  xr, x2v + 1, orow + 1472, lane); // 2x1->1
    run_path<5, 3, 5, 2>(cg + 420,  xr, x2v + 1, orow + 4800, lane); // 2x1->2
    run_path<5, 3, 7, 2>(cg + 495,  xr, x2v + 1, orow + 6016, lane); // 2x1->3
    run_path<5, 5, 1, 2>(cg + 600,  xr, x2v + 4, orow + 256,  lane); // 2x2->0
    run_path<5, 5, 3, 2>(cg + 625,  xr, x2v + 4, orow + 2048, lane); // 2x2->1
    run_path<5, 5, 5, 2>(cg + 700,  xr, x2v + 4, orow + 3840, lane); // 2x2->2
    run_path<5, 5, 7, 2>(cg + 825,  xr, x2v + 4, orow + 6464, lane); // 2x2->3
    run_path<5, 5, 9, 2>(cg + 1000, xr, x2v + 4, orow + 6912, lane); // 2x2->4
  }
}

// ===========================================================================
// Path table (enumeration order of the reference's nested loops)
// ===========================================================================
__device__ const int PATH_L1[19] = {0,0,0, 1,1,1,1,1,1,1, 2,2,2,2,2,2,2,2,2};
__device__ const int PATH_L2[19] = {0,1,2, 0,1,1,1,2,2,2, 0,1,1,1,2,2,2,2,2};
__device__ const int PATH_L3[19] = {0,1,2, 1,0,1,2,1,2,3, 2,1,2,3,0,1,2,3,4};
// dense cg sizes: 1,9,25, 9,9,27,45, 45,75,105, 25, 45,75,105, 25,75,125,175,225 (total 1225)
__device__ const int PATH_CGOFF[19] = {0,1,10, 35,44,53,80, 125,170,245,
                                       350,375,420,495, 600,625,700,825,1000};

// ===========================================================================
// Setup kernel: e3nn real Wigner-3j tables computed on device (fp64), exactly
// mirroring the reference (_su2_cg + complex change of basis + Fro-norm +
// sqrt(2*l3+1) "component" scaling).  One block per path; graph-capture safe.
// ===========================================================================
__device__ __forceinline__ double dfact(int n) {
  double r = 1.0;
  for (int i = 2; i <= n; ++i) r *= (double)i;
  return r;
}

__device__ double su2cg(int j1, int j2, int j3, int m1, int m2, int m3) {
  if (m1 + m2 != m3) return 0.0;
  int vmin = max(max(-j1 + j2 + m3, -j1 + m1), 0);
  int vmax = min(min(j2 + j3 + m1, j3 - j1 + j2), j3 + m3);
  if (vmax < vmin) return 0.0;
  double C = sqrt((2.0 * j3 + 1.0) * dfact(j3 + j1 - j2) * dfact(j3 - j1 + j2) *
                  dfact(j1 + j2 - j3) * dfact(j3 + m3) * dfact(j3 - m3) /
                  (dfact(j1 + j2 + j3 + 1) * dfact(j1 - m1) * dfact(j1 + m1) *
                   dfact(j2 - m2) * dfact(j2 + m2)));
  double S = 0.0;
  for (int v = vmin; v <= vmax; ++v) {
    double sgn = ((v + j2 + m2) & 1) ? -1.0 : 1.0;
    S += sgn / dfact(v) * dfact(j2 + j3 + m1 - v) * dfact(j1 - m1 + v) /
         (dfact(j3 - j1 + j2 - v) * dfact(j3 + m3 - v) * dfact(v + j1 - j2 - m3));
  }
  return C * S;
}

// Q[a][b] of _change_basis_real_to_complex(l), including the (-1j)^l factor.
__device__ void qelem(int l, int a, int b, double& qr, double& qi) {
  const double s = 0.70710678118654752440;
  int m = a - l;
  double r = 0.0, i = 0.0;
  if (m < 0) {
    if (b == l - m) r = s;            // col l+|m|
    else if (b == l + m) i = -s;      // col l-|m|
  } else if (m == 0) {
    if (b == l) r = 1.0;
  } else {
    double sg = (m & 1) ? -1.0 : 1.0;
    if (b == l + m) r = sg * s;
    else if (b == l - m) i = sg * s;
  }
  switch (l & 3) {                    // multiply by (-i)^l
    case 0: qr = r;  qi = i;  break;
    case 1: qr = i;  qi = -r; break;
    case 2: qr = -r; qi = -i; break;
    default: qr = -i; qi = r; break;
  }
}

__global__ __launch_bounds__(128) void cg_setup_kernel(float* __restrict__ cg) {
  __shared__ double ssum[128];
  __shared__ double sscale;
  const int p = blockIdx.x;
  const int l1 = PATH_L1[p], l2 = PATH_L2[p], l3 = PATH_L3[p];
  const int d1 = 2 * l1 + 1, d2 = 2 * l2 + 1, d3 = 2 * l3 + 1;
  const int nel = d1 * d2 * d3;
  const int off = PATH_CGOFF[p];

  double mysum = 0.0;
  for (int e = threadIdx.x; e < nel; e += 128) {
    const int a1 = e / (d2 * d3);
    const int a2 = (e / d3) % d2;
    const int a3 = e % d3;
    // out[a1,a2,a3] = Re( sum_{i,k,n} Q1[i][a1] Q2[k][a2] conj(Q3[n][a3]) C[i,k,n] )
    double ar = 0.0;
    for (int i1 = 0; i1 < d1; ++i1) {
      double q1r, q1i; qelem(l1, i1, a1, q1r, q1i);
      if (q1r == 0.0 && q1i == 0.0) continue;
      for (int k2 = 0; k2 < d2; ++k2) {
        double q2r, q2i; qelem(l2, k2, a2, q2r, q2i);
        if (q2r == 0.0 && q2i == 0.0) continue;
        const double pr = q1r * q2r - q1i * q2i;
        const double pi = q1r * q2i + q1i * q2r;
        for (int n3 = 0; n3 < d3; ++n3) {
          const double c = su2cg(l1, l2, l3, i1 - l1, k2 - l2, n3 - l3);
          if (c == 0.0) continue;
          double q3r, q3i; qelem(l3, n3, a3, q3r, q3i);
          ar += c * (pr * q3r + pi * q3i);   // Re( p * conj(q3) )
        }
      }
    }
    cg[off + e] = (float)ar;   // unnormalized
    mysum += ar * ar;
  }
  ssum[threadIdx.x] = mysum;
  __syncthreads();
  for (int s = 64; s > 0; s >>= 1) {
    if ((int)threadIdx.x < s) ssum[threadIdx.x] += ssum[threadIdx.x + s];
    __syncthreads();
  }
  if (threadIdx.x == 0) sscale = sqrt((double)(2 * l3 + 1)) / sqrt(ssum[0]);
  __syncthreads();
  const float scl = (float)sscale;
  for (int e = threadIdx.x; e < nel; e += 128) cg[off + e] *= scl;
}

// ===========================================================================
extern "C" void kernel_launch(void* const* d_in, const int* in_sizes, int n_in,
                              void* d_out, int out_size, void* d_ws, size_t ws_size,
                              hipStream_t stream) {
  (void)n_in; (void)out_size; (void)ws_size;
  const float* in1 = (const float*)d_in[0];
  const float* in2 = (const float*)d_in[1];
  float* out = (float*)d_out;
  float* cg = (float*)d_ws;                      // 1225 floats of scratch
  const int nrows = in_sizes[0] / 832;           // 16384

  cg_setup_kernel<<<dim3(19), dim3(128), 0, stream>>>(cg);
  ftp_kernel<<<dim3((nrows + 7) / 8), dim3(256), 0, stream>>>(in1, in2, cg, out, nrows);
}